// CLIPTTA_44796508897394
// MI455X (gfx1250) — compile-verified
//
#include <hip/hip_runtime.h>
#include <hip/hip_bf16.h>

#define NC 1000
#define NM 50
#define ND 1024
#define NL 196
#define NTILES 13              // ceil(196/16)

typedef float v2f __attribute__((ext_vector_type(2)));
typedef float v8f __attribute__((ext_vector_type(8)));

// ---------------------------------------------------------------------------
// Kernel 1: all the small math. 1 block, 1024 threads (32 waves).
// Produces: ws_att[0..1023] = normalized attention vector
//           ws_meta[0]=pseudo, ws_meta[1]=slot, ws_meta[2]=do_write
// ---------------------------------------------------------------------------
__global__ void __launch_bounds__(1024)
clipta_prep(const float* __restrict__ init_pred,
            const float* __restrict__ loc,        // [L, D]
            const float* __restrict__ text,       // [C, D]
            const float* __restrict__ ent_mem,    // [C, M]
            const int*   __restrict__ cnt,        // [C, 1]
            float* __restrict__ ws_att,
            int*   __restrict__ ws_meta)
{
    const int tid = threadIdx.x;
    __shared__ float s_red[1024];
    __shared__ int   s_idx[1024];
    __shared__ float s_ent[1024];
    __shared__ float s_t[ND];
    __shared__ float s_cos[NTILES * 16];
    __shared__ float s_w[NL];

    // ---- argmax / value / entropy over init_pred[0:1000] ----
    float p = (tid < NC) ? init_pred[tid] : -1.0f;           // probs are >= 0
    float e = (tid < NC) ? -(p * logf(p + 1e-8f)) : 0.0f;
    s_red[tid] = p; s_idx[tid] = tid; s_ent[tid] = e;
    for (int s = 512; s > 0; s >>= 1) {
        __syncthreads();
        if (tid < s) {
            if (s_red[tid + s] > s_red[tid]) {               // strict >: first max wins
                s_red[tid] = s_red[tid + s];
                s_idx[tid] = s_idx[tid + s];
            }
            s_ent[tid] += s_ent[tid + s];
        }
    }
    __syncthreads();
    const int   pseudo = s_idx[0];
    const float value  = s_red[0];
    const float ent    = s_ent[0];
    __syncthreads();

    // ---- stage t = text_feat[pseudo] into LDS ----
    s_t[tid] = text[(size_t)pseudo * ND + tid];
    __syncthreads();

    // ---- cos[l] = dot(loc[l], t) via v_wmma_f32_16x16x4_f32 ----
    // Wave w (< 13) handles rows 16w..16w+15.
    // A (16x4 f32): lanes 0-15 hold M=lane with K=0 (v0), K=1 (v1);
    //               lanes 16-31 hold M=lane-16 with K=2 (v0), K=3 (v1).
    // B (4x16 f32): broadcast t over all 16 columns; lane-half K split
    //               mirrors A, so each lane feeds t[k] matching its A K's.
    // Rows >= 196 of the last tile are CLAMPED to row 195: their cos values
    // (s_cos[196..207]) are dead (softmax only reads tid < 196), and the
    // clamp keeps the loads uniform -> no exec-mask save/restore in the loop.
    {
        const int wave = tid >> 5;
        const int lane = tid & 31;
        if (wave < NTILES) {
            const int mbase = wave * 16;
            int       mrow  = mbase + (lane & 15);
            if (mrow >= NL) mrow = NL - 1;                   // dead rows, any value ok
            const int koff  = (lane >= 16) ? 2 : 0;
            const v2f* arow = (const v2f*)(loc + (size_t)mrow * ND);   // 8B aligned
            v8f acc = {0.f, 0.f, 0.f, 0.f, 0.f, 0.f, 0.f, 0.f};
            for (int c4 = 0; c4 < ND; c4 += 4) {
                const int k0 = c4 + koff;                    // even -> v2f index k0/2
                v2f b;
                b.x = s_t[k0];
                b.y = s_t[k0 + 1];
                const v2f a = arow[k0 >> 1];
                acc = __builtin_amdgcn_wmma_f32_16x16x4_f32(
                        false, a, false, b, (short)0, acc, false, false);
            }
            // C/D layout: vgpr r, lanes 0-15 -> M=r ; lanes 16-31 -> M=r+8.
            if (lane == 0) {
                #pragma unroll
                for (int r = 0; r < 8; ++r) s_cos[mbase + r] = acc[r];
            }
            if (lane == 16) {
                #pragma unroll
                for (int r = 0; r < 8; ++r) s_cos[mbase + 8 + r] = acc[r];
            }
        }
    }
    __syncthreads();

    // ---- softmax over 196 logits (cos * 50) ----
    const float lg = (tid < NL) ? s_cos[tid] * 50.0f : -INFINITY;
    s_red[tid] = lg;
    for (int s = 512; s > 0; s >>= 1) {
        __syncthreads();
        if (tid < s) s_red[tid] = fmaxf(s_red[tid], s_red[tid + s]);
    }
    __syncthreads();
    const float mx = s_red[0];
    __syncthreads();
    const float ex = (tid < NL) ? expf(lg - mx) : 0.0f;
    s_red[tid] = ex;
    for (int s = 512; s > 0; s >>= 1) {
        __syncthreads();
        if (tid < s) s_red[tid] += s_red[tid + s];
    }
    __syncthreads();
    const float wsum = s_red[0];
    if (tid < NL) s_w[tid] = ex / wsum;
    __syncthreads();

    // ---- att[d] = sum_l w[l] * loc[l][d], then normalize ----
    float a = 0.0f;
    for (int l = 0; l < NL; ++l)
        a += s_w[l] * loc[(size_t)l * ND + tid];            // coalesced per l
    s_red[tid] = a * a;
    for (int s = 512; s > 0; s >>= 1) {
        __syncthreads();
        if (tid < s) s_red[tid] += s_red[tid + s];
    }
    __syncthreads();
    const float nrm = sqrtf(s_red[0]);
    ws_att[tid] = a / nrm;

    // ---- slot / do_write decision ----
    if (tid == 0) {
        float emax = -INFINITY; int worst = 0;
        for (int j = 0; j < NM; ++j) {
            const float v = ent_mem[(size_t)pseudo * NM + j];
            if (v > emax) { emax = v; worst = j; }          // first max wins
        }
        const int  cn         = cnt[pseudo];
        const bool full       = (cn >= NM);
        const bool replace_ok = (ent < emax);
        const bool do_write   = (value > 0.0f) && (full ? replace_ok : true);
        int slot = full ? worst : (cn < 0 ? 0 : (cn > NM - 1 ? NM - 1 : cn));
        ws_meta[0] = pseudo;
        ws_meta[1] = slot;
        ws_meta[2] = do_write ? 1 : 0;
    }
}

// ---------------------------------------------------------------------------
// Kernel 2: bandwidth kernel. One 4 KB output row per block, staged through
// LDS with the CDNA5 async global<->LDS data movers (ASYNCcnt-tracked).
// Each lane moves 16 B; each wave only stores what it loaded, so the only
// ordering needed is s_wait_asynccnt 0 between the load and the store.
// ---------------------------------------------------------------------------
__global__ void __launch_bounds__(256)
clipta_copy(const float* __restrict__ img_mem,   // [C, M, D]
            const float* __restrict__ loc_mem,   // [C, M, D]
            float* __restrict__ out)             // [C, 2M, D]
{
    __shared__ __align__(16) float buf[ND];      // 4 KB staging
    const int row = blockIdx.x;                  // 0 .. C*2M-1
    const int c   = row / (2 * NM);
    const int r   = row - c * (2 * NM);

    const float* src = (r < NM)
        ? (img_mem + ((size_t)c * NM + r)        * ND)
        : (loc_mem + ((size_t)c * NM + (r - NM)) * ND);

    const float* gsrc = src + (size_t)threadIdx.x * 4;                 // 16 B / lane
    float*       gdst = out + (size_t)row * ND + (size_t)threadIdx.x * 4;
    const unsigned loff = (unsigned)(size_t)(&buf[0]) + threadIdx.x * 16u;

    asm volatile("global_load_async_to_lds_b128 %0, %1, off"
                 :: "v"(loff), "v"(gsrc) : "memory");
    asm volatile("s_wait_asynccnt 0" ::: "memory");
    asm volatile("global_store_async_from_lds_b128 %0, %1, off"
                 :: "v"(gdst), "v"(loff) : "memory");
    // s_endpgm performs an implicit wait-idle, draining the async stores.
}

// ---------------------------------------------------------------------------
// Kernel 3: conditional row replacement at (pseudo, slot) / (pseudo, M+slot).
// ---------------------------------------------------------------------------
__global__ void __launch_bounds__(256)
clipta_fixup(const float* __restrict__ gfeat,    // [1, D]
             const float* __restrict__ ws_att,   // [D]
             const int*   __restrict__ meta,
             float* __restrict__ out)
{
    if (meta[2] == 0) return;
    const int pseudo = meta[0];
    const int slot   = meta[1];
    const float* src = (blockIdx.x == 0) ? gfeat : ws_att;
    const size_t row = (size_t)pseudo * (2 * NM) + slot + (blockIdx.x == 0 ? 0 : NM);

    const float4* s4 = (const float4*)src;
    float4*       d4 = (float4*)(out + row * ND);
    d4[threadIdx.x] = s4[threadIdx.x];
}

// ---------------------------------------------------------------------------
extern "C" void kernel_launch(void* const* d_in, const int* in_sizes, int n_in,
                              void* d_out, int out_size, void* d_ws, size_t ws_size,
                              hipStream_t stream)
{
    const float* init_pred = (const float*)d_in[0];   // [1, C]
    const float* gfeat     = (const float*)d_in[1];   // [1, D]
    const float* locfeat   = (const float*)d_in[2];   // [1, L, D]
    const float* text      = (const float*)d_in[3];   // [C, D]
    const float* img_mem   = (const float*)d_in[4];   // [C, M, D]
    const float* loc_mem   = (const float*)d_in[5];   // [C, M, D]
    const float* ent_mem   = (const float*)d_in[6];   // [C, M]
    const int*   cnt       = (const int*)  d_in[7];   // [C, 1]
    float*       out       = (float*)d_out;           // [C, 2M, D]

    float* ws_att  = (float*)d_ws;                    // 1024 floats
    int*   ws_meta = (int*)((char*)d_ws + ND * sizeof(float));

    clipta_prep<<<1, 1024, 0, stream>>>(init_pred, locfeat, text, ent_mem, cnt,
                                        ws_att, ws_meta);
    clipta_copy<<<NC * 2 * NM, 256, 0, stream>>>(img_mem, loc_mem, out);
    clipta_fixup<<<2, 256, 0, stream>>>(gfeat, ws_att, ws_meta, out);
}